// ChanAttNorm_36240934044377
// MI455X (gfx1250) — compile-verified
//
#include <hip/hip_runtime.h>
#include <hip/hip_bf16.h>
#include <math.h>

typedef float v2f __attribute__((ext_vector_type(2)));
typedef float v8f __attribute__((ext_vector_type(8)));
typedef int v4i __attribute__((vector_size(16)));
typedef __attribute__((address_space(1))) v4i gas_v4i;   // global (device) int4
typedef __attribute__((address_space(3))) v4i las_v4i;   // LDS (shared) int4

#if defined(__has_builtin)
#if __has_builtin(__builtin_amdgcn_global_load_async_to_lds_b128) && \
    __has_builtin(__builtin_amdgcn_s_wait_asynccnt)
#define USE_ASYNC_LDS 1
#endif
#endif
#ifndef USE_ASYNC_LDS
#define USE_ASYNC_LDS 0
#endif

#define EPSV 1e-7f
#define BN 8
#define CN 128
#define NN 16
#define HWN 16384
#define SPN 32                 // split-K slices per batch
#define KSN (HWN / SPN)        // 512 spatial positions per block
#define CHK 32                 // k per LDS chunk (double-buffered)
#define NCHK (KSN / CHK)       // 16 chunks
#define LSTR 36                // padded LDS row stride (conflict-free frag loads, 16B aligned)

// ---------------- Pass 1: partial Gram matrices G[b] = x x^T and column sums u ----------------
__global__ __launch_bounds__(256) void gram_partial(const float* __restrict__ x,
                                                    float* __restrict__ Gpart,
                                                    float* __restrict__ upart) {
  __shared__ float tile[2][CN * LSTR];            // 2 x 18.4 KB double buffer
  const int bb = blockIdx.x / SPN;
  const int sl = blockIdx.x % SPN;
  const int t = threadIdx.x;
  const int wave = t >> 5;
  const int lane = t & 31;
  const int lj = lane & 15;
  const int kh = (lane < 16) ? 0 : 2;             // K offset per half-wave (f32 WMMA layout)
  const float* xb = x + ((size_t)bb * CN) * HWN + sl * KSN;

  v8f acc[8] = {};                                // 8 16x16 tiles: rows wave*16.., cols rb*16..
  float usum = 0.f;

  // Stage one chunk (128 rows x 32 floats) into tile[buf]; 4 x B128 per thread.
  auto stage = [&](int buf, int ck) {
#pragma unroll
    for (int i = 0; i < 4; ++i) {
      const int idx = t + i * 256;
      const int c0 = idx >> 3;
      const int sg = (idx & 7) << 2;
#if USE_ASYNC_LDS
      __builtin_amdgcn_global_load_async_to_lds_b128(
          (gas_v4i*)(xb + (size_t)c0 * HWN + ck * CHK + sg),
          (las_v4i*)(&tile[buf][c0 * LSTR + sg]), 0, 0);
#else
      float4 v = *(const float4*)(xb + (size_t)c0 * HWN + ck * CHK + sg);
      *(float4*)(&tile[buf][c0 * LSTR + sg]) = v;
#endif
    }
  };

#if USE_ASYNC_LDS
  stage(0, 0);
  stage(1, 1);
#endif
  for (int ck = 0; ck < NCHK; ++ck) {
    const int buf = ck & 1;
#if USE_ASYNC_LDS
    // This wave's 4 oldest async ops (current buffer) must have retired; in-order completion.
    __builtin_amdgcn_s_wait_asynccnt(4);
    __syncthreads();
#else
    __syncthreads();
    stage(buf, ck);
    __syncthreads();
#endif
    if (t < CN) {
#pragma unroll 8
      for (int k = 0; k < CHK; ++k) usum += tile[buf][t * LSTR + k];
    }
    for (int ks = 0; ks < CHK; ks += 4) {
      // A-frag (own row block) and B-frags (all row blocks) share the same lane layout.
      v2f afr = *(const v2f*)(&tile[buf][(wave * 16 + lj) * LSTR + ks + kh]);
      v2f frag[8];
#pragma unroll
      for (int rb = 0; rb < 8; ++rb)
        frag[rb] = *(const v2f*)(&tile[buf][(rb * 16 + lj) * LSTR + ks + kh]);
#pragma unroll
      for (int rb = 0; rb < 8; ++rb)
        acc[rb] = __builtin_amdgcn_wmma_f32_16x16x4_f32(false, afr, false, frag[rb],
                                                        (short)0, acc[rb], false, false);
    }
#if USE_ASYNC_LDS
    __syncthreads();                              // everyone done reading tile[buf]
    if (ck + 2 < NCHK) stage(buf, ck + 2);        // prefetch chunk ck+2 into the freed buffer
#endif
  }
  float* gp = Gpart + (size_t)(bb * SPN + sl) * (CN * CN);
  const int rbase = wave * 16 + ((lane < 16) ? 0 : 8);
#pragma unroll
  for (int rb = 0; rb < 8; ++rb)
#pragma unroll
    for (int g = 0; g < 8; ++g)
      gp[(size_t)(rbase + g) * CN + rb * 16 + lj] = acc[rb][g];
  if (t < CN) upart[(bb * SPN + sl) * CN + t] = usum;
}

// ---------------- Pass 2: deterministic split-K reduction ----------------
__global__ __launch_bounds__(256) void reduce_gu(const float* __restrict__ Gpart,
                                                 const float* __restrict__ upart,
                                                 float* __restrict__ Gfull,
                                                 float* __restrict__ ufull) {
  const int bb = blockIdx.x >> 3;
  const int pp = blockIdx.x & 7;
  const int t = threadIdx.x;
  for (int i = t; i < 512; i += 256) {
    int off = pp * 2048 + i * 4;
    float4 s = make_float4(0.f, 0.f, 0.f, 0.f);
    for (int sl = 0; sl < SPN; ++sl) {
      float4 v = *(const float4*)(Gpart + (size_t)(bb * SPN + sl) * (CN * CN) + off);
      s.x += v.x; s.y += v.y; s.z += v.z; s.w += v.w;
    }
    *(float4*)(Gfull + (size_t)bb * (CN * CN) + off) = s;
  }
  if (pp == 0 && t < CN) {
    float s = 0.f;
    for (int sl = 0; sl < SPN; ++sl) s += upart[(bb * SPN + sl) * CN + t];
    ufull[bb * CN + t] = s;
  }
}

// ---------------- Pass 3: all attention math from G and u (one block per batch) ----------------
__global__ __launch_bounds__(256) void attn_small(
    const float* __restrict__ Gfull, const float* __restrict__ ufull,
    const float* __restrict__ Wsa, const float* __restrict__ bsa,
    const float* __restrict__ Wr,  const float* __restrict__ br,
    const float* __restrict__ kn,
    const float* __restrict__ Wo,  const float* __restrict__ bo,
    const float* __restrict__ alpha, const float* __restrict__ sigma,
    float* __restrict__ Aout, float* __restrict__ Bout) {
  __shared__ float u_s[CN], wsum_s[CN], v_s[CN];
  __shared__ float dsa_s[CN], dr_s[CN], do_s[CN];
  __shared__ float wusa_s[CN], wur_s[CN], wuo_s[CN];
  __shared__ float mf_s[CN], reg_s[CN], nxr_s[CN], s1_s[CN], s2_s[CN];
  __shared__ float Z_s[NN * CN], rn_s[NN * CN], att_s[NN * CN];
  __shared__ float knu_s[NN], nxn_s[NN], inv_s[NN], fminv_s[NN];
  __shared__ float bsum_s, mx_s, sm_s;
  const int bb = blockIdx.x;
  const int t = threadIdx.x;
  const float* Gf = Gfull + (size_t)bb * CN * CN;

  if (t < CN) {
    u_s[t] = ufull[bb * CN + t];
    float s = 0.f;
    for (int j = 0; j < CN; ++j) s += Wsa[j * CN + t];
    wsum_s[t] = s;
  }
  if (t == 0) {
    float s = 0.f;
    for (int j = 0; j < CN; ++j) s += bsa[j];
    bsum_s = s;
  }
  __syncthreads();
  // v = G * wsum  (threads 0..127), quadratic diagonals for Wsa / Wr (0..127 / 128..255)
  if (t < CN) {
    float s = 0.f;
    for (int q = 0; q < CN; ++q) s += Gf[t * CN + q] * wsum_s[q];
    v_s[t] = s;
  }
  {
    const int j = t & 127;
    const float* Wm = (t < CN) ? Wsa : Wr;
    float d = 0.f, wu = 0.f;
    for (int p = 0; p < CN; ++p) {
      float wp = Wm[j * CN + p];
      wu += wp * u_s[p];
      float yp = 0.f;
      for (int q = 0; q < CN; ++q) yp += Gf[p * CN + q] * Wm[j * CN + q];
      d += wp * yp;
    }
    if (t < CN) { dsa_s[j] = d; wusa_s[j] = wu; }
    else        { dr_s[j]  = d; wur_s[j]  = wu; }
  }
  __syncthreads();
  // Wo diagonal (0..127) in parallel with Z = kn @ G (128..255)
  if (t < CN) {
    const int j = t;
    float d = 0.f, wu = 0.f;
    for (int p = 0; p < CN; ++p) {
      float wp = Wo[j * CN + p];
      wu += wp * u_s[p];
      float yp = 0.f;
      for (int q = 0; q < CN; ++q) yp += Gf[p * CN + q] * Wo[j * CN + q];
      d += wp * yp;
    }
    do_s[j] = d; wuo_s[j] = wu;
  } else {
    const int tl = t - CN;
    const int i = tl >> 3;
    const int q0 = (tl & 7) * 16;
    for (int qq = 0; qq < 16; ++qq) {
      const int q = q0 + qq;
      float s = 0.f;
      for (int p = 0; p < CN; ++p) s += kn[i * CN + p] * Gf[p * CN + q];
      Z_s[i * CN + q] = s;
    }
  }
  __syncthreads();
  if (t < CN) {
    const int j = t;
    float wsumu = 0.f;
    for (int q = 0; q < CN; ++q) wsumu += wsum_s[q] * u_s[q];
    float t1 = 0.f;
    for (int p = 0; p < CN; ++p) t1 += v_s[p] * Wsa[j * CN + p];
    t1 += bsum_s * wusa_s[j] + bsa[j] * wsumu + bsa[j] * bsum_s * (float)HWN;
    float t2 = dsa_s[j] + 2.f * bsa[j] * wusa_s[j] + bsa[j] * bsa[j] * (float)HWN;
    mf_s[j] = (t1 - t2) * (1.f / (float)CN);
    nxr_s[j] = sqrtf(dr_s[j] + 2.f * br[j] * wur_s[j] + br[j] * br[j] * (float)HWN);
    s1_s[j] = wuo_s[j] + bo[j] * (float)HWN;
    s2_s[j] = do_s[j] + 2.f * bo[j] * wuo_s[j] + bo[j] * bo[j] * (float)HWN;
  } else if (t < CN + NN) {
    const int i = t - CN;
    float ku = 0.f, nx = 0.f;
    for (int q = 0; q < CN; ++q) {
      ku += kn[i * CN + q] * u_s[q];
      nx += Z_s[i * CN + q] * kn[i * CN + q];
    }
    knu_s[i] = ku;
    nxn_s[i] = sqrtf(nx);
  }
  __syncthreads();
  if (t == 0) {
    float mx = mf_s[0];
    for (int j = 1; j < CN; ++j) mx = fmaxf(mx, mf_s[j]);
    mx_s = mx;
  }
  __syncthreads();
  if (t < CN) reg_s[t] = expf(mf_s[t] - mx_s);
  __syncthreads();
  if (t == 0) {
    float s = 0.f;
    for (int j = 0; j < CN; ++j) s += reg_s[j];
    sm_s = s;
  }
  __syncthreads();
  if (t < CN) reg_s[t] /= sm_s;
  __syncthreads();
  {
    const int i = t >> 4;
    const int jg = t & 15;
    float ac = fminf(fmaxf(alpha[i], 0.f), 1.f);
    for (int jj = 0; jj < 8; ++jj) {
      const int j = jg * 8 + jj;
      float rel = 0.f;
      for (int q = 0; q < CN; ++q) rel += Z_s[i * CN + q] * Wr[j * CN + q];
      rel += br[j] * knu_s[i];
      rn_s[i * CN + j] = rel / (nxn_s[i] * nxr_s[j] + EPSV) + ac * reg_s[j];
    }
  }
  __syncthreads();
  if (t < CN) {                                   // softmax over n per channel column
    const int j = t;
    float mx = rn_s[j];
#pragma unroll
    for (int i = 1; i < NN; ++i) mx = fmaxf(mx, rn_s[i * CN + j]);
    float e[NN];
    float s = 0.f;
#pragma unroll
    for (int i = 0; i < NN; ++i) { e[i] = expf(rn_s[i * CN + j] - mx); s += e[i]; }
#pragma unroll
    for (int i = 0; i < NN; ++i) att_s[i * CN + j] = e[i] / s;
  }
  __syncthreads();
  if (t < NN) {
    const int i = t;
    float sa = 0.f, hot = 0.f, sb = 0.f;
    for (int j = 0; j < CN; ++j) {
      float a = att_s[i * CN + j];
      sa += a;
      hot += a * s1_s[j];
      sb += a * a * s2_s[j];
    }
    float cnt = sa * (float)HWN + EPSV;
    float fm = hot / cnt;
    float sq = sb - 2.f * fm * hot + fm * fm * (float)CN * (float)HWN;
    float fstd = sqrtf(sq / cnt);
    float iv = 1.f / (fstd + EPSV);
    inv_s[i] = iv;
    fminv_s[i] = fm * iv;
  }
  __syncthreads();
  if (t < CN) {
    float A = 0.f, Bv = 0.f;
#pragma unroll
    for (int i = 0; i < NN; ++i) {
      float a = att_s[i * CN + t];
      A += a * inv_s[i];
      Bv += a * fminv_s[i];
    }
    const float sg = sigma[0];
    Aout[bb * CN + t] = sg * A;
    Bout[bb * CN + t] = sg * Bv;
  }
}

// ---------------- Pass 4: out = x + (Wo x + bo)*A_c - B_c via WMMA, fused epilogue ----------------
__global__ __launch_bounds__(256) void conv_out(const float* __restrict__ x,
                                                const float* __restrict__ Wo,
                                                const float* __restrict__ bo,
                                                const float* __restrict__ Aco,
                                                const float* __restrict__ Bco,
                                                float* __restrict__ out) {
  __shared__ float wl[CN * CN];                   // wl[c*CN + j] = Wo[j][c]  (64 KB)
  const int bb = blockIdx.x >> 7;
  const int kb = blockIdx.x & 127;
  const int t = threadIdx.x;
  const int wave = t >> 5;
  const int lane = t & 31;
  const int lj = lane & 15;
  const int ch2 = (lane < 16) ? 0 : 2;
  for (int idx = t; idx < CN * CN; idx += 256) {
    int j = idx >> 7;
    int c = idx & 127;
    wl[c * CN + j] = Wo[idx];
  }
  __syncthreads();
  const int j0 = wave * 16;
  const int rbase = j0 + ((lane < 16) ? 0 : 8);
  const float* xb = x + (size_t)bb * CN * HWN;
  float* ob = out + (size_t)bb * CN * HWN;
  float boc[8], acf[8], bcf[8];
#pragma unroll
  for (int g = 0; g < 8; ++g) {
    boc[g] = bo[rbase + g];
    acf[g] = Aco[bb * CN + rbase + g];
    bcf[g] = Bco[bb * CN + rbase + g];
  }
  for (int kt = 0; kt < 8; ++kt) {
    const int k0 = kb * 128 + kt * 16;
    v8f acc;
#pragma unroll
    for (int g = 0; g < 8; ++g) acc[g] = boc[g];  // bias pre-loaded into C operand
    for (int cp = 0; cp < CN; cp += 4) {
      v2f a, b;
      a.x = wl[(cp + ch2) * CN + j0 + lj];
      a.y = wl[(cp + ch2 + 1) * CN + j0 + lj];
      b.x = xb[(size_t)(cp + ch2) * HWN + k0 + lj];
      b.y = xb[(size_t)(cp + ch2 + 1) * HWN + k0 + lj];
      acc = __builtin_amdgcn_wmma_f32_16x16x4_f32(false, a, false, b, (short)0, acc, false, false);
    }
#pragma unroll
    for (int g = 0; g < 8; ++g) {
      const size_t off = (size_t)(rbase + g) * HWN + k0 + lj;
      ob[off] = xb[off] + acc[g] * acf[g] - bcf[g];
    }
  }
}

// ---------------- Pass 5: orthogonality loss (tiny) ----------------
__global__ __launch_bounds__(256) void orth_kernel(const float* __restrict__ kn,
                                                   float* __restrict__ loss_out) {
  __shared__ float sym_s[NN * NN];
  __shared__ float nrm_s[NN];
  __shared__ float red_s[NN * NN];
  const int t = threadIdx.x;
  if (t < NN * NN) {
    const int i = t >> 4, j = t & 15;
    float s = 0.f;
    for (int p = 0; p < CN; ++p) s += kn[i * CN + p] * kn[j * CN + p];
    sym_s[t] = s;
  }
  __syncthreads();
  if (t < NN) nrm_s[t] = sqrtf(sym_s[t * NN + t]);
  __syncthreads();
  if (t < NN * NN) {
    const int i = t >> 4, j = t & 15;
    float l = sym_s[t] / (nrm_s[i] * nrm_s[j] + EPSV) - ((i == j) ? 1.f : 0.f);
    red_s[t] = l * l;
  }
  __syncthreads();
  if (t == 0) {
    float s = 0.f;
    for (int q = 0; q < NN * NN; ++q) s += red_s[q];
    loss_out[0] = 0.1f * logf(s + 1.f);
  }
}

extern "C" void kernel_launch(void* const* d_in, const int* in_sizes, int n_in,
                              void* d_out, int out_size, void* d_ws, size_t ws_size,
                              hipStream_t stream) {
  (void)in_sizes; (void)n_in; (void)out_size; (void)ws_size;
  const float* x     = (const float*)d_in[0];
  const float* Wsa   = (const float*)d_in[1];
  const float* bsa   = (const float*)d_in[2];
  const float* Wr    = (const float*)d_in[3];
  const float* br    = (const float*)d_in[4];
  const float* kn    = (const float*)d_in[5];
  const float* Wo    = (const float*)d_in[6];
  const float* bo    = (const float*)d_in[7];
  const float* alpha = (const float*)d_in[8];
  const float* sigma = (const float*)d_in[9];
  float* out = (float*)d_out;

  float* w = (float*)d_ws;
  float* Gpart = w;                                   // 8*32*128*128 floats (16 MB)
  float* upart = Gpart + (size_t)BN * SPN * CN * CN;  // 8*32*128
  float* Gfull = upart + (size_t)BN * SPN * CN;       // 8*128*128
  float* ufull = Gfull + (size_t)BN * CN * CN;        // 8*128
  float* Aco   = ufull + (size_t)BN * CN;             // 8*128
  float* Bco   = Aco   + (size_t)BN * CN;             // 8*128

  gram_partial<<<BN * SPN, 256, 0, stream>>>(x, Gpart, upart);
  reduce_gu<<<BN * 8, 256, 0, stream>>>(Gpart, upart, Gfull, ufull);
  attn_small<<<BN, 256, 0, stream>>>(Gfull, ufull, Wsa, bsa, Wr, br, kn, Wo, bo,
                                     alpha, sigma, Aco, Bco);
  conv_out<<<BN * 128, 256, 0, stream>>>(x, Wo, bo, Aco, Bco, out);
  orth_kernel<<<1, 256, 0, stream>>>(kn, out + (size_t)BN * CN * HWN);
}